// ContrastiveLoss_20512763806185
// MI455X (gfx1250) — compile-verified
//
#include <hip/hip_runtime.h>

#define WAVE 32

typedef __attribute__((ext_vector_type(2))) float v2f;
typedef __attribute__((ext_vector_type(8))) float v8f;

// Full-wave (32-lane) fp32 sum using the CDNA5 fp32 WMMA (V_WMMA_F32_16X16X4_F32).
// A (16x4 fp32) layout: lane L holds row L%16, K = 2*(L/16) in a[0], K+1 in a[1].
// With a[0] = lane partial, a[1] = 0 and B = ones(4x16):
//   D[i][j] = p[lane i] + p[lane i+16]  (replicated across all 16 columns).
// C/D layout: lane L holds column L%16, rows (L<16 ? 0..7 : 8..15) in c[0..7].
// Summing c[0..7] gives rows 0..7 (lanes 0-15) / rows 8..15 (lanes 16-31) of the
// column sum; adding the xor-16 partner yields the full 32-lane sum in every lane.
// Requires EXEC all-ones at the call site (guaranteed by uniform control flow).
__device__ __forceinline__ float wave_sum_f32(float x) {
  v2f a;    a[0] = x;    a[1] = 0.0f;
  v2f ones; ones[0] = 1.0f; ones[1] = 1.0f;
  v8f c = {};
  c = __builtin_amdgcn_wmma_f32_16x16x4_f32(
      /*neg_a=*/false, a, /*neg_b=*/false, ones,
      /*c_mod=*/(short)0, c, /*reuse_a=*/false, /*reuse_b=*/false);
  float s = ((c[0] + c[1]) + (c[2] + c[3])) + ((c[4] + c[5]) + (c[6] + c[7]));
  s += __shfl_xor(s, 16, 32);
  return s;
}

// One lane per sampled pair. Gathers two 64B descriptors (4x b128 each),
// computes squared L2 distance (optionally hinged), reduces per-wave via WMMA,
// and writes one fp32 partial per wave (no atomics -> deterministic).
// Wave-level loop bounds are uniform; tail lanes are masked arithmetically so
// EXEC stays all-ones for the WMMA.
template <bool HINGE>
__global__ void gather_dist_kernel(const float* __restrict__ descA,
                                   const float* __restrict__ descB,
                                   const int* __restrict__ idxA,
                                   const int* __restrict__ idxB,
                                   float* __restrict__ partials,
                                   int M, int N, float margin) {
  const int lane        = threadIdx.x & (WAVE - 1);
  const int waveInBlk   = threadIdx.x >> 5;
  const int wavesPerBlk = blockDim.x >> 5;
  const int b           = blockIdx.y;                       // batch index
  const int gwx         = blockIdx.x * wavesPerBlk + waveInBlk; // wave id in batch slice
  const int nwx         = gridDim.x * wavesPerBlk;              // waves per batch slice

  const size_t idxBase  = (size_t)b * (size_t)M;
  const size_t descBase = (size_t)b * (size_t)N * 4;  // in float4 units (D=16 -> 4 float4)
  const float4* __restrict__ a4 = (const float4*)descA;
  const float4* __restrict__ b4 = (const float4*)descB;

  float partial = 0.0f;
  for (int s0 = gwx * WAVE; s0 < M; s0 += nwx * WAVE) {  // uniform per wave
    const int  m     = s0 + lane;
    const bool valid = (m < M);
    const int  mc    = valid ? m : (M - 1);

    const int ia = idxA[idxBase + mc];
    const int ib = idxB[idxBase + mc];
    const float4* pa = a4 + descBase + (size_t)ia * 4;
    const float4* pb = b4 + descBase + (size_t)ib * 4;

    float dist = 0.0f;
#pragma unroll
    for (int k = 0; k < 4; ++k) {
      const float4 va = pa[k];
      const float4 vb = pb[k];
      const float dx = va.x - vb.x;
      const float dy = va.y - vb.y;
      const float dz = va.z - vb.z;
      const float dw = va.w - vb.w;
      dist = __builtin_fmaf(dx, dx, dist);
      dist = __builtin_fmaf(dy, dy, dist);
      dist = __builtin_fmaf(dz, dz, dist);
      dist = __builtin_fmaf(dw, dw, dist);
    }
    const float contrib = HINGE ? fmaxf(margin - dist, 0.0f) : dist;
    partial += valid ? contrib : 0.0f;
  }

  const float wsum = wave_sum_f32(partial);   // EXEC all-ones here
  if (lane == 0) {
    partials[((size_t)b * gridDim.x + blockIdx.x) * wavesPerBlk + waveInBlk] = wsum;
  }
}

// Single-block deterministic final reduction: strided per-thread accumulation
// (fixed order), WMMA wave sums, LDS combine, scale, and write the 3 outputs:
// out[0]=contrastive_sum, out[1]=match_sum, out[2]=nonmatch_sum.
__global__ void reduce_finalize_kernel(const float* __restrict__ pm, int nm,
                                       const float* __restrict__ pn, int nn,
                                       float* __restrict__ out,
                                       float invM, float invN) {
  __shared__ float smem[2][8];
  const int tid  = threadIdx.x;
  const int lane = tid & (WAVE - 1);
  const int wv   = tid >> 5;

  float am = 0.0f, an = 0.0f;
  for (int i = tid; i < nm; i += blockDim.x) am += pm[i];
  for (int i = tid; i < nn; i += blockDim.x) an += pn[i];

  am = wave_sum_f32(am);   // control flow reconverged: EXEC all-ones
  an = wave_sum_f32(an);

  if (lane == 0) { smem[0][wv] = am; smem[1][wv] = an; }
  __syncthreads();

  if (tid == 0) {
    const int nw = blockDim.x >> 5;
    float sm = 0.0f, sn = 0.0f;
    for (int w = 0; w < nw; ++w) { sm += smem[0][w]; sn += smem[1][w]; }
    const float mloss = sm * invM;
    const float nloss = sn * invN;  // NON_MATCH_WEIGHT == 1.0
    out[0] = mloss + nloss;
    out[1] = mloss;
    out[2] = nloss;
  }
}

extern "C" void kernel_launch(void* const* d_in, const int* in_sizes, int n_in,
                              void* d_out, int out_size, void* d_ws, size_t ws_size,
                              hipStream_t stream) {
  const float* outA   = (const float*)d_in[0];
  const float* outB   = (const float*)d_in[1];
  const int*   matchA = (const int*)d_in[2];
  const int*   matchB = (const int*)d_in[3];
  const int*   nonA   = (const int*)d_in[4];
  const int*   nonB   = (const int*)d_in[5];

  const int Bb = 4;
  const int D  = 16;
  const int N  = in_sizes[0] / (Bb * D);   // 307200
  const int Mm = in_sizes[2] / Bb;         // 5000
  const int Mn = in_sizes[4] / Bb;         // 50000

  const int threads     = 256;             // 8 waves per block (wave32)
  const int wavesPerBlk = threads / WAVE;

  int blkM = (Mm + threads - 1) / threads; // 20
  int blkN = (Mn + threads - 1) / threads; // 196

  // Clamp grids so the per-wave partial slots fit the workspace
  // (grid-stride loops inside the kernels keep full coverage).
  const size_t maxFloats = ws_size / sizeof(float);
  while ((size_t)Bb * (size_t)(blkM + blkN) * wavesPerBlk > maxFloats && blkN > 1) blkN >>= 1;
  while ((size_t)Bb * (size_t)(blkM + blkN) * wavesPerBlk > maxFloats && blkM > 1) blkM >>= 1;

  const int nm = Bb * blkM * wavesPerBlk;
  const int nn = Bb * blkN * wavesPerBlk;
  float* pm = (float*)d_ws;
  float* pn = pm + nm;

  gather_dist_kernel<false><<<dim3(blkM, Bb), threads, 0, stream>>>(
      outA, outB, matchA, matchB, pm, Mm, N, 0.0f);
  gather_dist_kernel<true><<<dim3(blkN, Bb), threads, 0, stream>>>(
      outA, outB, nonA, nonB, pn, Mn, N, 0.5f);
  reduce_finalize_kernel<<<1, threads, 0, stream>>>(
      pm, nm, pn, nn, (float*)d_out, 1.0f / (float)Mm, 1.0f / (float)Mn);
}